// GIN_Regress_66760971649441
// MI455X (gfx1250) — compile-verified
//
#include <hip/hip_runtime.h>
#include <hip/hip_bf16.h>

typedef __attribute__((ext_vector_type(2))) float v2f;
typedef __attribute__((ext_vector_type(8))) float v8f;

#define HF 10  // hidden feature width

static __device__ __forceinline__ v8f wmma_f32_k4(v2f a, v2f b, v8f c) {
  // D = A(16x4 f32) x B(4x16 f32) + C(16x16 f32)
  return __builtin_amdgcn_wmma_f32_16x16x4_f32(false, a, false, b, (short)0, c,
                                               false, false);
}

// ------------- pack weights into zero-padded 16x16 tiles / 16-vec biases ----
// blockIdx.x = matrix slot m: 0 -> preW(16x10); 1+2l -> convW1[l]; 2+2l -> convW2[l]
__global__ void pack_weights(const float* __restrict__ preW,
                             const float* __restrict__ preb,
                             const float* __restrict__ cW1,
                             const float* __restrict__ cb1,
                             const float* __restrict__ cW2,
                             const float* __restrict__ cb2,
                             float* __restrict__ wpad,   // [(1+2L)][16][16]
                             float* __restrict__ bpad) { // [(1+2L)][16]
  const int m = blockIdx.x;
  const int t = threadIdx.x;  // 0..255
  const int k = t >> 4, n = t & 15;
  const float* Wsrc;
  const float* bsrc;
  int Krows;
  if (m == 0) {
    Wsrc = preW; bsrc = preb; Krows = 16;
  } else {
    const int l = (m - 1) >> 1;
    if ((m - 1) & 1) { Wsrc = cW2 + l * HF * HF; bsrc = cb2 + l * HF; }
    else             { Wsrc = cW1 + l * HF * HF; bsrc = cb1 + l * HF; }
    Krows = HF;
  }
  wpad[m * 256 + t] = (k < Krows && n < HF) ? Wsrc[k * HF + n] : 0.0f;
  if (t < 16) bpad[m * 16 + t] = (t < HF) ? bsrc[t] : 0.0f;
}

// ---------------- pre_mp: h = x @ preW + preb  ([N,16]@[16,10]) --------------
__global__ void pre_mp_wmma(const float* __restrict__ x,
                            const float* __restrict__ Wp,  // [16][16] padded
                            const float* __restrict__ bp,  // [16] padded
                            float* __restrict__ h, int N) {
  const int wave   = threadIdx.x >> 5;
  const int waveId = blockIdx.x * (blockDim.x >> 5) + wave;
  const int node0  = waveId << 4;
  if (node0 >= N) return;  // wave-uniform: EXEC all-ones for WMMA
  const int lane = threadIdx.x & 31;
  const int lh   = lane >> 4;
  const int ln   = lane & 15;

  const float bv = bp[ln];  // unconditional (padded)
  v8f c = {bv, bv, bv, bv, bv, bv, bv, bv};

  const int arow = min(node0 + ln, N - 1);  // clamp: bad rows discarded later
#pragma unroll
  for (int s = 0; s < 4; ++s) {  // K = 16
    const int k = 4 * s + 2 * lh;  // even -> 8B aligned pair
    const float2 av = *(const float2*)(x + (size_t)arow * 16 + k);
    v2f a, b;
    a.x = av.x;
    a.y = av.y;
    b.x = Wp[k * 16 + ln];
    b.y = Wp[(k + 1) * 16 + ln];
    c = wmma_f32_k4(a, b, c);
  }
  if (node0 + 16 <= N) {  // wave-uniform full-tile fast path
    if (ln < HF) {
#pragma unroll
      for (int v = 0; v < 8; ++v) h[(size_t)(node0 + v + 8 * lh) * HF + ln] = c[v];
    }
  } else {
#pragma unroll
    for (int v = 0; v < 8; ++v) {
      const int row = node0 + v + 8 * lh;
      if (row < N && ln < HF) h[(size_t)row * HF + ln] = c[v];
    }
  }
}

// ---------------- utility: vectorized copy / zero ---------------------------
__global__ void copy_f32(const float* __restrict__ src, float* __restrict__ dst,
                         long n) {
  const long i      = (long)blockIdx.x * blockDim.x + threadIdx.x;
  const long stride = (long)gridDim.x * blockDim.x;
  const long n4     = n >> 2;
  const float4* s4  = (const float4*)src;
  float4* d4        = (float4*)dst;
  for (long j = i; j < n4; j += stride) d4[j] = s4[j];
  for (long j = (n4 << 2) + i; j < n; j += stride) dst[j] = src[j];
}

__global__ void zero_f32(float* __restrict__ p, int n) {
  const int i = blockIdx.x * blockDim.x + threadIdx.x;
  if (i < n) p[i] = 0.0f;
}

// ---------------- edge scatter: agg[dst] += h[src] (agg pre-seeded with h) --
__global__ void scatter_edges(const int* __restrict__ ei,
                              const float* __restrict__ h,
                              float* __restrict__ agg, int E) {
  const int e = blockIdx.x * blockDim.x + threadIdx.x;
  if (e >= E) return;
  const int src = ei[e];
  const int dst = ei[E + e];
  const float2* hs = (const float2*)(h + (size_t)src * HF);  // 8B aligned
  float* ad = agg + (size_t)dst * HF;
#pragma unroll
  for (int f = 0; f < 5; ++f) {
    const float2 v = hs[f];
    atomicAdd(&ad[2 * f],     v.x);  // non-returning global_atomic_add_f32
    atomicAdd(&ad[2 * f + 1], v.y);
  }
}

// ---------------- GIN MLP: hout = relu(z@W1+b1)@W2+b2, + BN stat partials ---
__global__ void gin_mlp_wmma(const float* __restrict__ z,    // [N,10] (+pad)
                             const float* __restrict__ W1p,  // [16][16] padded
                             const float* __restrict__ b1p,  // [16] padded
                             const float* __restrict__ W2p,  // [16][16] padded
                             const float* __restrict__ b2p,  // [16] padded
                             float* __restrict__ hout,
                             float* __restrict__ bnSum,
                             float* __restrict__ bnSqSum, int N, int doStats) {
  __shared__ float t[8][16][17];  // per-wave intermediate tile (+1 pad)
  const int wave   = threadIdx.x >> 5;
  const int waveId = blockIdx.x * (blockDim.x >> 5) + wave;
  const int node0  = waveId << 4;
  if (node0 >= N) return;  // wave-uniform
  const int lane = threadIdx.x & 31;
  const int lh   = lane >> 4;
  const int ln   = lane & 15;
  const int arow = min(node0 + ln, N - 1);  // clamp; bad rows discarded later

  // ---- first matmul: K = 10 padded to 12 (3 K4 steps) ----
  const float bv1 = b1p[ln];
  v8f c = {bv1, bv1, bv1, bv1, bv1, bv1, bv1, bv1};
#pragma unroll
  for (int s = 0; s < 3; ++s) {
    const int k = 4 * s + 2 * lh;  // even
    // unconditional load (z buffer has >=16 floats of tail padding);
    // zero K-pad lanes via selects (fold away for s=0,1 after unroll)
    const float2 zv = *(const float2*)(z + (size_t)arow * HF + k);
    v2f a, b;
    a.x = (k     < HF) ? zv.x : 0.0f;
    a.y = (k + 1 < HF) ? zv.y : 0.0f;
    b.x = W1p[k * 16 + ln];
    b.y = W1p[(k + 1) * 16 + ln];
    c = wmma_f32_k4(a, b, c);
  }
  // relu, park D tile in LDS to re-read in A-matrix layout (same-wave DS ops
  // execute in order; compiler barrier prevents reordering)
#pragma unroll
  for (int v = 0; v < 8; ++v) t[wave][v + 8 * lh][ln] = fmaxf(c[v], 0.0f);
  asm volatile("" ::: "memory");

  // ---- second matmul: K = 16 (cols 10..15 of t are exactly zero) ----
  const float bv2 = b2p[ln];
  v8f d = {bv2, bv2, bv2, bv2, bv2, bv2, bv2, bv2};
#pragma unroll
  for (int s = 0; s < 4; ++s) {
    const int k = 4 * s + 2 * lh;
    v2f a, b;
    a.x = t[wave][ln][k];
    a.y = t[wave][ln][k + 1];
    b.x = W2p[k * 16 + ln];
    b.y = W2p[(k + 1) * 16 + ln];
    d = wmma_f32_k4(a, b, d);
  }

  float s1 = 0.0f, s2 = 0.0f;
  if (node0 + 16 <= N) {  // wave-uniform full-tile fast path
#pragma unroll
    for (int v = 0; v < 8; ++v) {
      const float val = d[v];
      s1 += val;
      s2 += val * val;
    }
    if (ln < HF) {
#pragma unroll
      for (int v = 0; v < 8; ++v)
        hout[(size_t)(node0 + v + 8 * lh) * HF + ln] = d[v];
    }
  } else {
#pragma unroll
    for (int v = 0; v < 8; ++v) {
      const int row = node0 + v + 8 * lh;
      if (row < N) {
        const float val = d[v];
        if (ln < HF) hout[(size_t)row * HF + ln] = val;
        s1 += val;
        s2 += val * val;
      }
    }
  }
  if (doStats && ln < HF) {
    atomicAdd(&bnSum[ln], s1);
    atomicAdd(&bnSqSum[ln], s2);
  }
}

// ---------------- BN: fold into per-channel affine, then apply --------------
__global__ void bn_finalize(const float* __restrict__ sums,
                            const float* __restrict__ sqs,
                            const float* __restrict__ gamma,
                            const float* __restrict__ beta,
                            float* __restrict__ bnA, float* __restrict__ bnB,
                            int N) {
  const int f = threadIdx.x;
  if (f >= HF) return;
  const float invN = 1.0f / (float)N;
  const float mean = sums[f] * invN;
  const float var  = sqs[f] * invN - mean * mean;  // biased variance
  const float a    = gamma[f] * rsqrtf(var + 1e-5f);
  bnA[f] = a;
  bnB[f] = beta[f] - mean * a;
}

__global__ void bn_apply(float* __restrict__ h, const float* __restrict__ bnA,
                         const float* __restrict__ bnB, int N) {
  const int i = blockIdx.x * blockDim.x + threadIdx.x;
  if (i >= N) return;
  float2* hp = (float2*)(h + (size_t)i * HF);  // 8B aligned
#pragma unroll
  for (int f = 0; f < 5; ++f) {
    float2 v = hp[f];
    v.x = v.x * bnA[2 * f]     + bnB[2 * f];
    v.y = v.y * bnA[2 * f + 1] + bnB[2 * f + 1];
    hp[f] = v;
  }
}

// ---------------- global mean pool + post MLP -------------------------------
__global__ void pool_nodes(const float* __restrict__ h,
                           const int* __restrict__ batch,
                           float* __restrict__ psum, float* __restrict__ pcnt,
                           int N) {
  const int i = blockIdx.x * blockDim.x + threadIdx.x;
  if (i >= N) return;
  const int g = batch[i];
  const float2* hs = (const float2*)(h + (size_t)i * HF);
  float* pd = psum + (size_t)g * HF;
#pragma unroll
  for (int f = 0; f < 5; ++f) {
    const float2 v = hs[f];
    atomicAdd(&pd[2 * f],     v.x);
    atomicAdd(&pd[2 * f + 1], v.y);
  }
  atomicAdd(&pcnt[g], 1.0f);
}

__global__ void post_mlp(const float* __restrict__ psum,
                         const float* __restrict__ pcnt,
                         const float* __restrict__ W1,
                         const float* __restrict__ b1,
                         const float* __restrict__ W2,
                         const float* __restrict__ b2,
                         float* __restrict__ out, int G) {
  const int g = blockIdx.x * blockDim.x + threadIdx.x;
  if (g >= G) return;
  const float inv = 1.0f / fmaxf(pcnt[g], 1.0f);
  float m[HF];
#pragma unroll
  for (int f = 0; f < HF; ++f) m[f] = psum[(size_t)g * HF + f] * inv;
  float o = b2[0];
#pragma unroll
  for (int j = 0; j < HF; ++j) {
    float acc = b1[j];
#pragma unroll
    for (int f = 0; f < HF; ++f) acc += m[f] * W1[f * HF + j];
    o += fmaxf(acc, 0.0f) * W2[j];
  }
  out[g] = o;
}

// ---------------------------------------------------------------------------
extern "C" void kernel_launch(void* const* d_in, const int* in_sizes, int n_in,
                              void* d_out, int out_size, void* d_ws,
                              size_t ws_size, hipStream_t stream) {
  const float* x      = (const float*)d_in[0];
  const int*   ei     = (const int*)d_in[1];
  const int*   batch  = (const int*)d_in[2];
  const float* preW   = (const float*)d_in[3];
  const float* preb   = (const float*)d_in[4];
  const float* convW1 = (const float*)d_in[5];
  const float* convb1 = (const float*)d_in[6];
  const float* convW2 = (const float*)d_in[7];
  const float* convb2 = (const float*)d_in[8];
  const float* bn_g   = (const float*)d_in[9];
  const float* bn_b   = (const float*)d_in[10];
  const float* postW1 = (const float*)d_in[11];
  const float* postb1 = (const float*)d_in[12];
  const float* postW2 = (const float*)d_in[13];
  const float* postb2 = (const float*)d_in[14];
  float* out = (float*)d_out;

  const int N = in_sizes[0] / 16;
  const int E = in_sizes[1] / 2;
  const int G = out_size;
  const int L = in_sizes[5] / (HF * HF);
  const int NMAT = 1 + 2 * L;  // padded weight slots

  // workspace layout (256B-aligned slices); ws is poisoned -> zero what we use
  auto align256 = [](size_t v) { return (v + 255) & ~(size_t)255; };
  char* w = (char*)d_ws;
  size_t off = 0;
  float* h   = (float*)(w + off); off = align256(off + ((size_t)N * HF + 16) * 4);
  float* agg = (float*)(w + off); off = align256(off + ((size_t)N * HF + 16) * 4);
  float* wpad = (float*)(w + off); off = align256(off + (size_t)NMAT * 256 * 4);
  float* bpad = (float*)(w + off); off = align256(off + (size_t)NMAT * 16 * 4);
  float* bnStats = (float*)(w + off); off = align256(off + 32 * 4);
  float* bnSum = bnStats;
  float* bnSq  = bnStats + 16;
  float* bnA = (float*)(w + off); off = align256(off + 16 * 4);
  float* bnB = (float*)(w + off); off = align256(off + 16 * 4);
  float* pool = (float*)(w + off); off = align256(off + (size_t)G * (HF + 1) * 4);
  float* psum = pool;                   // [G,HF]
  float* pcnt = pool + (size_t)G * HF;  // [G]
  (void)ws_size; (void)n_in;

  const int  waves = (N + 15) / 16;
  const int  wgrid = (waves + 7) / 8;  // 256 threads = 8 wave32 per block
  const long nh    = (long)N * HF;

  pack_weights<<<NMAT, 256, 0, stream>>>(preW, preb, convW1, convb1, convW2,
                                         convb2, wpad, bpad);
  pre_mp_wmma<<<wgrid, 256, 0, stream>>>(x, wpad, bpad, h, N);

  for (int l = 0; l < L; ++l) {
    const int doStats = (l < L - 1) ? 1 : 0;
    copy_f32<<<1024, 256, 0, stream>>>(h, agg, nh);  // agg := h (GIN self term)
    scatter_edges<<<(E + 255) / 256, 256, 0, stream>>>(ei, h, agg, E);
    if (doStats) zero_f32<<<1, 32, 0, stream>>>(bnStats, 32);
    gin_mlp_wmma<<<wgrid, 256, 0, stream>>>(
        agg, wpad + (size_t)(1 + 2 * l) * 256, bpad + (size_t)(1 + 2 * l) * 16,
        wpad + (size_t)(2 + 2 * l) * 256, bpad + (size_t)(2 + 2 * l) * 16, h,
        bnSum, bnSq, N, doStats);
    if (doStats) {
      bn_finalize<<<1, 32, 0, stream>>>(bnSum, bnSq, bn_g + (size_t)l * HF,
                                        bn_b + (size_t)l * HF, bnA, bnB, N);
      bn_apply<<<(N + 255) / 256, 256, 0, stream>>>(h, bnA, bnB, N);
    }
  }

  zero_f32<<<(G * (HF + 1) + 255) / 256, 256, 0, stream>>>(pool, G * (HF + 1));
  pool_nodes<<<(N + 255) / 256, 256, 0, stream>>>(h, batch, psum, pcnt, N);
  post_mlp<<<(G + 255) / 256, 256, 0, stream>>>(psum, pcnt, postW1, postb1,
                                                postW2, postb2, out, G);
}